// MemoryRepresentation_49426483642737
// MI455X (gfx1250) — compile-verified
//
#include <hip/hip_runtime.h>

#define N_SLOTS 100000
#define D 256
#define NUM_HOPS 3
#define N_TILES (N_SLOTS / 16)   // 6250 exactly

typedef __attribute__((ext_vector_type(2))) float v2f;
typedef __attribute__((ext_vector_type(8))) float v8f;

// Order-preserving float->uint encoding so we can use integer atomicMax for
// the softmax running max (handles negative scores).
__device__ __forceinline__ unsigned enc_f32(float f) {
  unsigned u = __float_as_uint(f);
  return (u & 0x80000000u) ? ~u : (u | 0x80000000u);
}
__device__ __forceinline__ float dec_f32(unsigned u) {
  return __uint_as_float((u & 0x80000000u) ? (u ^ 0x80000000u) : ~u);
}

// u[j] = sum_k B[j,k] * query[k]   (one block of 256 threads)
__global__ void memnet_init_u(const float* __restrict__ B,
                              const float* __restrict__ q,
                              float* __restrict__ u) {
  int j = threadIdx.x;
  float acc = 0.f;
#pragma unroll 8
  for (int k = 0; k < D; ++k) acc += B[j * D + k] * q[k];
  u[j] = acc;
}

// v[k] = sum_j A[j,k] * u[j]; also reset per-hop stats and wsum accumulator.
__global__ void memnet_prep_hop(const float* __restrict__ A,
                                const float* __restrict__ u,
                                float* __restrict__ v,
                                unsigned* __restrict__ maxEnc,
                                float* __restrict__ Z,
                                float* __restrict__ wsum) {
  int k = threadIdx.x;
  float acc = 0.f;
#pragma unroll 8
  for (int j = 0; j < D; ++j) acc += A[j * D + k] * u[j];  // coalesced in k
  v[k] = acc;
  wsum[k] = 0.f;
  if (k == 0) { maxEnc[0] = 0u; Z[0] = 0.f; }
}

// scores[i] = dot(memory[i], v) via V_WMMA_F32_16X16X4_F32.
// One wave per 16-row tile. B matrix = v broadcast across all 16 columns,
// so every column of D equals the score vector for the tile's rows.
__global__ void memnet_scores_wmma(const float* __restrict__ mem,
                                   const float* __restrict__ v,
                                   float* __restrict__ scores,
                                   unsigned* __restrict__ maxEnc) {
  int wid  = (blockIdx.x * blockDim.x + threadIdx.x) >> 5;
  int lane = threadIdx.x & 31;
  if (wid >= N_TILES) return;  // wave-uniform: EXEC stays all-ones for WMMA

  int row0 = wid * 16;
  int m    = lane & 15;            // row within tile (A layout)
  int kh   = (lane >> 4) << 1;     // lanes 0-15 -> K 0,1 ; lanes 16-31 -> K 2,3

  const float* arow = mem + (size_t)(row0 + m) * D + kh;
  const float* bvec = v + kh;

  v8f c0 = {};
  v8f c1 = {};
#pragma unroll 4
  for (int k0 = 0; k0 < D; k0 += 8) {
    v2f a0 = *(const v2f*)(arow + k0);
    v2f b0 = *(const v2f*)(bvec + k0);
    v2f a1 = *(const v2f*)(arow + k0 + 4);
    v2f b1 = *(const v2f*)(bvec + k0 + 4);
    c0 = __builtin_amdgcn_wmma_f32_16x16x4_f32(false, a0, false, b0,
                                               (short)0, c0, false, false);
    c1 = __builtin_amdgcn_wmma_f32_16x16x4_f32(false, a1, false, b1,
                                               (short)0, c1, false, false);
  }
  v8f c = c0 + c1;

  // C/D layout: lane 0 holds D[0..7][0], lane 16 holds D[8..15][0].
  float mx = -3.4e38f;
  if ((lane & 15) == 0) {
    int rbase = row0 + ((lane >> 4) << 3);
#pragma unroll
    for (int r = 0; r < 8; ++r) {
      float s = c[r];
      scores[rbase + r] = s;
      mx = fmaxf(mx, s);
    }
  }
  float other = __shfl_xor(mx, 16, 32);
  mx = fmaxf(mx, other);
  if (lane == 0) atomicMax(maxEnc, enc_f32(mx));
}

// e_i = exp(scores_i - gmax); Z += sum e_i; wsum[t] += sum_i e_i * mem[i][t].
// One block per 256-row chunk; thread t owns column t (coalesced rows).
__global__ void memnet_attend(const float* __restrict__ mem,
                              const float* __restrict__ scores,
                              const unsigned* __restrict__ maxEnc,
                              float* __restrict__ wsum,
                              float* __restrict__ Z) {
  __shared__ float e[256];
  __shared__ float red[256];
  int t = threadIdx.x;
  int row0 = blockIdx.x * 256;
  int nrows = (N_SLOTS - row0 < 256) ? (N_SLOTS - row0) : 256;

  float gmax = dec_f32(maxEnc[0]);
  float myE = 0.f;
  if (t < nrows) myE = __expf(scores[row0 + t] - gmax);
  e[t] = myE;
  red[t] = myE;
  __syncthreads();
  for (int s = 128; s > 0; s >>= 1) {
    if (t < s) red[t] += red[t + s];
    __syncthreads();
  }

  float acc = 0.f;
  const float* p = mem + (size_t)row0 * D + t;
#pragma unroll 8
  for (int r = 0; r < nrows; ++r) acc += e[r] * p[(size_t)r * D];

  atomicAdd(&wsum[t], acc);
  if (t == 0) atomicAdd(Z, red[0]);
}

// u[j] += (1/Z) * sum_k C[j,k] * wsum[k]; optionally emit final output.
__global__ void memnet_update_u(const float* __restrict__ C,
                                const float* __restrict__ wsum,
                                const float* __restrict__ Z,
                                float* __restrict__ u,
                                float* __restrict__ out,
                                int write_out) {
  int j = threadIdx.x;
  float invZ = 1.f / Z[0];
  float acc = 0.f;
#pragma unroll 8
  for (int k = 0; k < D; ++k) acc += C[j * D + k] * wsum[k];
  float nu = u[j] + acc * invZ;
  u[j] = nu;
  if (write_out) out[j] = nu;
}

extern "C" void kernel_launch(void* const* d_in, const int* in_sizes, int n_in,
                              void* d_out, int out_size, void* d_ws, size_t ws_size,
                              hipStream_t stream) {
  const float* mem = (const float*)d_in[0];
  const float* q   = (const float*)d_in[1];
  const float* A   = (const float*)d_in[2];
  const float* B   = (const float*)d_in[3];
  const float* C   = (const float*)d_in[4];

  float* ws       = (float*)d_ws;
  float* u        = ws;                       // 256
  float* v        = ws + 256;                 // 256
  float* wsum     = ws + 512;                 // 256
  unsigned* maxEnc = (unsigned*)(ws + 768);   // 1 (encoded max)
  float* Z        = ws + 769;                 // 1
  float* scores   = ws + 1024;                // N_SLOTS

  memnet_init_u<<<1, 256, 0, stream>>>(B, q, u);

  const int scoreBlocks  = (N_TILES + 7) / 8;          // 8 waves/block
  const int attendBlocks = (N_SLOTS + 255) / 256;

  for (int h = 0; h < NUM_HOPS; ++h) {
    memnet_prep_hop<<<1, 256, 0, stream>>>(A, u, v, maxEnc, Z, wsum);
    memnet_scores_wmma<<<scoreBlocks, 256, 0, stream>>>(mem, v, scores, maxEnc);
    memnet_attend<<<attendBlocks, 256, 0, stream>>>(mem, scores, maxEnc, wsum, Z);
    memnet_update_u<<<1, 256, 0, stream>>>(C, wsum, Z, u, (float*)d_out,
                                           h == NUM_HOPS - 1 ? 1 : 0);
  }
}